// ReasoningFunction_28132035789121
// MI455X (gfx1250) — compile-verified
//
#include <hip/hip_runtime.h>
#include <math.h>
#include <stdint.h>

typedef __attribute__((ext_vector_type(2))) float v2f;
typedef __attribute__((ext_vector_type(8))) float v8f;

static constexpr int Dm = 512;   // model dim
static constexpr int Kr = 32;    // routed tokens per position

__device__ __forceinline__ float gelu_exact(float x) {
    // 0.5*x*(1+erf(x/sqrt(2))) — matches jax.nn.gelu(approximate=False)
    return 0.5f * x * (1.0f + erff(x * 0.70710678118654752f));
}

// Async-copy a 16x512 fp32 tile (32 KB) from global to the START of the
// dynamic-LDS block using CDNA5 GLOBAL_LOAD_ASYNC_TO_LDS_B128 (ASYNCcnt),
// then wait for this wave's async ops. Caller must __syncthreads() after.
__device__ __forceinline__ void async_tile_load_32k(const float4* __restrict__ src,
                                                    int tid, int nthreads) {
    const uint32_t lds_base = __builtin_amdgcn_groupstaticsize(); // dyn-LDS offset
    const int total = 16 * Dm / 4;                                // 2048 float4
    for (int i = tid; i < total; i += nthreads) {
        const uint64_t g = (uint64_t)(src + i);
        const uint32_t l = lds_base + (uint32_t)i * 16u;
        asm volatile("global_load_async_to_lds_b128 %0, %1, off"
                     :: "v"(l), "v"(g) : "memory");
    }
    asm volatile("s_wait_asynccnt 0x0" ::: "memory");
}

// ---------------------------------------------------------------------------
// Kernel A: hbase[b, e] = (token @ W_top)[b, e] + b_logic[e]
// grid = BT/16 blocks, 256 threads (8 waves). Wave w owns cols [w*64, w*64+64).
// ---------------------------------------------------------------------------
__global__ void __launch_bounds__(256)
hbase_kernel(const float* __restrict__ token,
             const float* __restrict__ Wtop,   // (512, 512)
             const float* __restrict__ b_logic,
             float* __restrict__ hbase) {
    extern __shared__ float lds[];                 // 16 x 512 A tile (32 KB)
    const int tid  = threadIdx.x;
    const int wave = tid >> 5, lane = tid & 31;
    const int hi   = lane >> 4, lo = lane & 15;

    async_tile_load_32k((const float4*)(token + (size_t)blockIdx.x * 16 * Dm),
                        tid, 256);
    __syncthreads();

    v8f c[4] = {};
    const int colbase = wave * 64;
    for (int kk = 0; kk < Dm; kk += 4) {
        const int k0 = kk + 2 * hi;                       // A frag: k = 2*hi + v
        v2f a = *(const v2f*)&lds[lo * Dm + k0];          // A[m=lo][k0..k0+1]
#pragma unroll
        for (int t = 0; t < 4; ++t) {
            const int e = colbase + t * 16 + lo;
            v2f bb;                                       // B frag: row = 2*hi + v
            bb.x = Wtop[(size_t)k0 * Dm + e];
            bb.y = Wtop[(size_t)(k0 + 1) * Dm + e];
            c[t] = __builtin_amdgcn_wmma_f32_16x16x4_f32(
                false, a, false, bb, (short)0, c[t], false, false);
        }
    }
#pragma unroll
    for (int t = 0; t < 4; ++t) {
        const int e = colbase + t * 16 + lo;
        const float bl = b_logic[e];
#pragma unroll
        for (int j = 0; j < 8; ++j) {
            const int m = j + 8 * hi;                     // C layout: row = j + 8*hi
            hbase[((size_t)blockIdx.x * 16 + m) * Dm + e] = c[t][j] + bl;
        }
    }
}

// ---------------------------------------------------------------------------
// Kernel B (fused): for one token b, over two 16-row k-tiles:
//   proj = routed[b, ktile] @ W_bot  (fp32 WMMA, A tile async-copied to LDS)
//   h    = gelu(proj + hbase[b])     (written back into the same LDS slab)
//   LN over D, then arg_summary[b] += softmax(sim[b])[k] * fused  (registers)
// grid = BT blocks, 512 threads (16 waves). Wave w owns cols [w*32, w*32+32).
// ---------------------------------------------------------------------------
__global__ void __launch_bounds__(512)
fused_kernel(const float* __restrict__ routed,
             const float* __restrict__ Wbot,   // (512, 512)
             const float* __restrict__ hbase,
             const float* __restrict__ sim_row,
             const float* __restrict__ gamma,
             const float* __restrict__ beta,
             float* __restrict__ argsum) {
    extern __shared__ float lds[];
    float* Ah   = lds;                   // 16*512: A tile, then reused for h
    float* psum = Ah   + 16 * Dm;        // 16*32 partial sums
    float* psq  = psum + 16 * 32;        // 16*32 partial sumsq
    float* mu   = psq  + 16 * 32;        // 16
    float* rstd = mu   + 16;             // 16
    float* wts  = rstd + 16;             // 32 softmax weights
    float* sbuf = wts  + Kr;             // 32

    const int b    = blockIdx.x;
    const int tid  = threadIdx.x;
    const int wave = tid >> 5, lane = tid & 31;
    const int hi   = lane >> 4, lo = lane & 15;
    const int colbase = wave * 32;

    if (tid < Kr) sbuf[tid] = sim_row[(size_t)b * Kr + tid];
    __syncthreads();
    if (tid == 0) {                      // K=32 softmax: trivial cost vs GEMM
        float mx = sbuf[0];
        for (int k = 1; k < Kr; ++k) mx = fmaxf(mx, sbuf[k]);
        float s = 0.f;
        for (int k = 0; k < Kr; ++k) { float e = expf(sbuf[k] - mx); wts[k] = e; s += e; }
        const float inv = 1.f / s;
        for (int k = 0; k < Kr; ++k) wts[k] *= inv;
    }

    float acc = 0.f;                     // thread owns output column e = tid

    for (int p = 0; p < 2; ++p) {
        __syncthreads();                 // prev pass done with Ah / wts visible
        // async copy routed rows [b*32 + p*16 .. +15] (contiguous 32 KB) to LDS
        async_tile_load_32k(
            (const float4*)(routed + ((size_t)b * Kr + p * 16) * Dm), tid, 512);
        __syncthreads();

        v8f c[2] = {};
        for (int kk = 0; kk < Dm; kk += 4) {
            const int k0 = kk + 2 * hi;
            v2f a = *(const v2f*)&Ah[lo * Dm + k0];
#pragma unroll
            for (int t = 0; t < 2; ++t) {
                const int e = colbase + t * 16 + lo;
                v2f bb;
                bb.x = Wbot[(size_t)k0 * Dm + e];
                bb.y = Wbot[(size_t)(k0 + 1) * Dm + e];
                c[t] = __builtin_amdgcn_wmma_f32_16x16x4_f32(
                    false, a, false, bb, (short)0, c[t], false, false);
            }
            if ((kk & 31) == 0 && kk + 32 < Dm)   // stream hint for W_bot (L2-hot)
                __builtin_prefetch(&Wbot[(size_t)(k0 + 32) * Dm + colbase + lo], 0, 1);
        }
        __syncthreads();                 // all waves finished reading A tile

        // h = gelu(proj + hbase) over the same LDS slab
#pragma unroll
        for (int t = 0; t < 2; ++t) {
            const int e = colbase + t * 16 + lo;
            const float hb = hbase[(size_t)b * Dm + e];
#pragma unroll
            for (int j = 0; j < 8; ++j) {
                const int m = j + 8 * hi;            // local k-row within tile
                Ah[m * Dm + e] = gelu_exact(c[t][j] + hb);
            }
        }
        __syncthreads();

        {   // LN stats: 16 rows x 32 chunks of 16 elements
            const int r = tid & 15, ch = tid >> 4;   // ch in 0..31
            const float* row = Ah + r * Dm + ch * 16;
            float s = 0.f, sq = 0.f;
#pragma unroll
            for (int i = 0; i < 16; ++i) { float x = row[i]; s += x; sq += x * x; }
            psum[r * 32 + ch] = s;
            psq [r * 32 + ch] = sq;
        }
        __syncthreads();
        if (tid < 16) {
            float s = 0.f, sq = 0.f;
            for (int ch = 0; ch < 32; ++ch) { s += psum[tid * 32 + ch]; sq += psq[tid * 32 + ch]; }
            const float m = s * (1.f / Dm);
            const float v = sq * (1.f / Dm) - m * m;
            mu[tid]   = m;
            rstd[tid] = rsqrtf(v + 1e-5f);
        }
        __syncthreads();

        {   // weighted accumulate: thread owns column e = tid (512 threads = D)
            const int e = tid;
            const float g = gamma[e], bt = beta[e];
            for (int k = 0; k < 16; ++k) {
                const float x = Ah[k * Dm + e];
                acc += wts[p * 16 + k] * ((x - mu[k]) * rstd[k] * g + bt);
            }
        }
    }
    argsum[(size_t)b * Dm + tid] = acc;
}

// ---------------------------------------------------------------------------
// Kernel C: gate = sigmoid(token @ Wg_top + argsum @ Wg_bot + b_gate)
// Effective K = 1024 GEMM in two stages sharing accumulators; 32 KB LDS.
// grid = BT/16 blocks, 256 threads (8 waves).
// ---------------------------------------------------------------------------
__global__ void __launch_bounds__(256)
gate_kernel(const float* __restrict__ token,
            const float* __restrict__ argsum,
            const float* __restrict__ Wgate,   // (1024, 512)
            const float* __restrict__ b_gate,
            float* __restrict__ gate) {
    extern __shared__ float lds[];                 // 16 x 512 A tile
    const int tid  = threadIdx.x;
    const int wave = tid >> 5, lane = tid & 31;
    const int hi   = lane >> 4, lo = lane & 15;
    const int colbase = wave * 64;

    v8f c[4] = {};
    for (int stage = 0; stage < 2; ++stage) {
        __syncthreads();
        const float* A = (stage == 0) ? token : argsum;
        const float* W = Wgate + (size_t)stage * Dm * Dm;
        async_tile_load_32k((const float4*)(A + (size_t)blockIdx.x * 16 * Dm),
                            tid, 256);
        __syncthreads();
        for (int kk = 0; kk < Dm; kk += 4) {
            const int k0 = kk + 2 * hi;
            v2f a = *(const v2f*)&lds[lo * Dm + k0];
#pragma unroll
            for (int t = 0; t < 4; ++t) {
                const int e = colbase + t * 16 + lo;
                v2f bb;
                bb.x = W[(size_t)k0 * Dm + e];
                bb.y = W[(size_t)(k0 + 1) * Dm + e];
                c[t] = __builtin_amdgcn_wmma_f32_16x16x4_f32(
                    false, a, false, bb, (short)0, c[t], false, false);
            }
        }
    }
#pragma unroll
    for (int t = 0; t < 4; ++t) {
        const int e = colbase + t * 16 + lo;
        const float bg = b_gate[e];
#pragma unroll
        for (int j = 0; j < 8; ++j) {
            const int m = j + 8 * hi;
            const float x = c[t][j] + bg;
            gate[((size_t)blockIdx.x * 16 + m) * Dm + e] = 1.f / (1.f + expf(-x));
        }
    }
}

// ---------------------------------------------------------------------------
extern "C" void kernel_launch(void* const* d_in, const int* in_sizes, int n_in,
                              void* d_out, int out_size, void* d_ws, size_t ws_size,
                              hipStream_t stream) {
    const float* token  = (const float*)d_in[0];   // (BT, 512)
    const float* routed = (const float*)d_in[1];   // (BT, 32, 512)
    const float* sim    = (const float*)d_in[2];   // (BT, 32)
    const float* Wlogic = (const float*)d_in[3];   // (1024, 512)
    const float* blogic = (const float*)d_in[4];   // (512,)
    const float* gamma  = (const float*)d_in[5];   // (512,)
    const float* beta   = (const float*)d_in[6];   // (512,)
    const float* Wgate  = (const float*)d_in[7];   // (1024, 512)
    const float* bgate  = (const float*)d_in[8];   // (512,)

    const int BT = in_sizes[0] / Dm;               // 8192
    float* out      = (float*)d_out;
    float* arg_out  = out;                         // (BT, 512)
    float* gate_out = out + (size_t)BT * Dm;       // (BT, 512)
    float* hbase    = (float*)d_ws;                // BT*512 floats = 16 MB scratch

    const size_t ldsA = (size_t)16 * Dm * sizeof(float);                       // 32 KB
    const size_t ldsB = (size_t)(16 * Dm + 16 * 32 * 2 + 16 + 16 + Kr + Kr) *
                        sizeof(float);                                         // ~37 KB

    hbase_kernel<<<BT / 16, 256, ldsA, stream>>>(token, Wlogic, blogic, hbase);
    fused_kernel<<<BT, 512, ldsB, stream>>>(routed, Wlogic + (size_t)Dm * Dm,
                                            hbase, sim, gamma, beta, arg_out);
    gate_kernel<<<BT / 16, 256, ldsA, stream>>>(token, arg_out, Wgate, bgate, gate_out);
}